// SelfAttention_20718922236033
// MI455X (gfx1250) — compile-verified
//
#include <hip/hip_runtime.h>

typedef __attribute__((ext_vector_type(2))) float v2f;
typedef __attribute__((ext_vector_type(8))) float v8f;

// Problem dims (fixed by the reference)
constexpr int Bn = 16;
constexpr int Ln = 2048;
constexpr int Dn = 512;

// ---------------------------------------------------------------------------
// Kernel 1: V[m,n] = sum_k x_bbox[m,k] * Wv[n,k] + bv[n]
//   M = 16 (batch), N = 512, K = 512.
// One wave32 per 16-wide N tile -> 32 waves total.
// Uses V_WMMA_F32_16X16X4_F32 so the projection stays fp32-exact.
//
// Per-lane layouts (CDNA5 ISA 7.12.2):
//   A (16x4 f32, 2 VGPRs):  lane&15 = M row; lanes 0-15 hold K={0,1},
//                           lanes 16-31 hold K={2,3}
//   B (4x16 f32, 2 VGPRs):  lane&15 = N col; same K split
//   C/D (16x16 f32, 8 VGPRs): element (M = r + 8*(lane>>4), N = lane&15)
// ---------------------------------------------------------------------------
__global__ __launch_bounds__(32) void proj_v_wmma(
    const float* __restrict__ xb,   // x_bbox  [16, 512]
    const float* __restrict__ Wv,   // Wv      [512, 512] row-major (n,k)
    const float* __restrict__ bv,   // bv      [512]
    float* __restrict__ V)          // out     [16, 512]
{
  const int lane = threadIdx.x;        // 0..31 (wave32)
  const int n0   = blockIdx.x * 16;    // N tile base
  const int q    = lane & 15;          // A M-row / B N-col / C N-col
  const int kh   = (lane >> 4) * 2;    // K sub-offset inside each 4-chunk

  const float* Arow = xb + q * Dn;            // A[m=q, :]
  const float* Brow = Wv + (n0 + q) * Dn;     // B[k, n=n0+q] == Wv[n0+q, k]

  v8f c = {};
  for (int kc = 0; kc < Dn; kc += 4) {
    v2f a = *(const v2f*)(Arow + kc + kh);    // 8B-aligned b64 load
    v2f b = *(const v2f*)(Brow + kc + kh);    // 8B-aligned b64 load
    // D = A x B + C, fp32 throughout
    c = __builtin_amdgcn_wmma_f32_16x16x4_f32(
        /*neg_a=*/false, a, /*neg_b=*/false, b,
        /*c_mod=*/(short)0, c, /*reuse_a=*/false, /*reuse_b=*/false);
  }

  const float bias = bv[n0 + q];
  const int   mh   = (lane >> 4) << 3;        // 0 or 8
#pragma unroll
  for (int r = 0; r < 8; ++r)
    V[(mh + r) * Dn + (n0 + q)] = c[r] + bias;
}

// ---------------------------------------------------------------------------
// Kernel 2: out[b, l, :] = V[b, :] for all l.  Pure store-bandwidth kernel:
// stage the 2KB row in LDS, then stream coalesced float4 (b128) stores.
// grid = Bn * (Ln/ROWS) = 4096 blocks, 256 threads each; 16KB written/block.
// ---------------------------------------------------------------------------
__global__ __launch_bounds__(256) void bcast_rows(
    const float* __restrict__ V,    // [16, 512]
    float* __restrict__ out)        // [16, 2048, 512]
{
  constexpr int ROWS = 8;
  constexpr int F4   = Dn / 4;                 // 128 float4 per row
  __shared__ float4 vrow[F4];                  // 2KB

  const int blocksPerB = Ln / ROWS;            // 256
  const int b  = blockIdx.x / blocksPerB;
  const int l0 = (blockIdx.x % blocksPerB) * ROWS;
  const int t  = threadIdx.x;

  if (t < F4) vrow[t] = ((const float4*)(V + b * Dn))[t];
  __syncthreads();

  float4* o = (float4*)(out + ((size_t)b * Ln + l0) * Dn);
#pragma unroll
  for (int i = t; i < ROWS * F4; i += 256)     // 4 b128 stores per thread
    o[i] = vrow[i & (F4 - 1)];
}

// ---------------------------------------------------------------------------
// Inputs (setup_inputs order):
//   0: x_img [16,2048,512]   (unused -- provably does not affect output)
//   1: x_img_bbox            (unused)
//   2: x_bbox [16,1,512]
//   3: Wq  4: bq  5: Wk  6: bk   (unused)
//   7: Wv [512,512]   8: bv [512]
// Output: [16, 2048, 512] fp32.  Workspace: 32KB for V.
// ---------------------------------------------------------------------------
extern "C" void kernel_launch(void* const* d_in, const int* in_sizes, int n_in,
                              void* d_out, int out_size, void* d_ws, size_t ws_size,
                              hipStream_t stream) {
  (void)in_sizes; (void)n_in; (void)out_size; (void)ws_size;

  const float* x_bbox = (const float*)d_in[2];
  const float* Wv     = (const float*)d_in[7];
  const float* bv     = (const float*)d_in[8];
  float*       out    = (float*)d_out;
  float*       V      = (float*)d_ws;          // 16*512 floats = 32KB

  proj_v_wmma<<<dim3(Dn / 16), dim3(32), 0, stream>>>(x_bbox, Wv, bv, V);
  bcast_rows<<<dim3(Bn * (Ln / 8)), dim3(256), 0, stream>>>(V, out);
}